// GNNModel_4964982194322
// MI455X (gfx1250) — compile-verified
//
#include <hip/hip_runtime.h>
#include <hip/hip_bf16.h>

typedef __attribute__((ext_vector_type(2))) float v2f;
typedef __attribute__((ext_vector_type(8))) float v8f;

#define HD   128          // hidden / channel dim
#define LSTR 132          // padded LDS row stride (dwords) for activations
#define WSTR 288          // packed weight k-pair stride: 256 data + 32 pad (bank-disjoint halves)
#define NGRAPH 256

// ---------------------------------------------------------------------------
// Edge scatter-add: agg[dst] += h[src].  One wave per edge, float4 per lane.
// ---------------------------------------------------------------------------
__global__ __launch_bounds__(256) void gin_scatter(const float* __restrict__ h,
                                                   const int* __restrict__ src,
                                                   const int* __restrict__ dst,
                                                   float* __restrict__ agg,
                                                   int E) {
    int gw   = (int)((blockIdx.x * blockDim.x + threadIdx.x) >> 5);
    int lane = (int)(threadIdx.x & 31);
    int nw   = (int)((gridDim.x * blockDim.x) >> 5);
    for (int e = gw; e < E; e += nw) {
        int s = src[e];
        int d = dst[e];
        float4 v = ((const float4*)h)[(size_t)s * (HD / 4) + lane];
        float* p = agg + (size_t)d * HD + lane * 4;
        unsafeAtomicAdd(p + 0, v.x);
        unsafeAtomicAdd(p + 1, v.y);
        unsafeAtomicAdd(p + 2, v.z);
        unsafeAtomicAdd(p + 3, v.w);
    }
}

// ---------------------------------------------------------------------------
// Cooperative load of a 128x128 weight matrix into K-pair-packed LDS layout:
//   element (k, col) -> wb[(k>>1)*WSTR + col*2 + (k&1)]
// so each B fragment ((k,col),(k+1,col)) is one aligned ds_load_b64.
// ---------------------------------------------------------------------------
__device__ __forceinline__ void load_weights_packed(const float* __restrict__ Wl,
                                                    float* wb, int tid) {
    for (int i = tid; i < (HD / 2) * (HD / 4); i += 256) {
        int kp = i >> 5, c4 = i & 31;
        float4 r0 = ((const float4*)Wl)[(size_t)(2 * kp + 0) * (HD / 4) + c4];
        float4 r1 = ((const float4*)Wl)[(size_t)(2 * kp + 1) * (HD / 4) + c4];
        float2* d = (float2*)(wb + (size_t)kp * WSTR + c4 * 8);
        d[0] = make_float2(r0.x, r1.x);
        d[1] = make_float2(r0.y, r1.y);
        d[2] = make_float2(r0.z, r1.z);
        d[3] = make_float2(r0.w, r1.w);
    }
}

// ---------------------------------------------------------------------------
// fp32 WMMA GEMM strip: acc[0..7] = 16 rows x 128 cols, K=128 in 32 steps of 4.
//   A: v2f from activation LDS (lanes 0-15: K=k0,k0+1; 16-31: K=k0+2,k0+3)
//   B: v2f ds_load_b64 from packed weight LDS (adjacent k-pair)
// ---------------------------------------------------------------------------
__device__ __forceinline__ void gemm_strip(const float* zb, const float* wb,
                                           int arow, int half, int lid,
                                           v8f acc[8]) {
    for (int kt = 0; kt < 32; ++kt) {
        int k0 = kt * 4 + 2 * half;
        v2f a = *(const v2f*)(zb + (size_t)arow * LSTR + k0);
        const float* wrow = wb + (size_t)(kt * 2 + half) * WSTR + lid * 2;
#pragma unroll
        for (int n = 0; n < 8; ++n) {
            v2f b = *(const v2f*)(wrow + n * 32);
            acc[n] = __builtin_amdgcn_wmma_f32_16x16x4_f32(
                false, a, false, b, (short)0, acc[n], false, false);
        }
    }
}

// ---------------------------------------------------------------------------
// Fused GIN layer MLP: z=(1+eps)*h+agg ; t=ReLU(z@W1+b1) ; out=t@W2+b2 (opt ReLU)
// One block = 128 rows.  8 waves; wave w owns rows [16w,16w+16).
// ---------------------------------------------------------------------------
__global__ __launch_bounds__(256) void gin_mlp(const float* __restrict__ hin,
                                               const float* __restrict__ agg,
                                               const float* __restrict__ W1,
                                               const float* __restrict__ b1,
                                               const float* __restrict__ W2,
                                               const float* __restrict__ b2,
                                               const float* __restrict__ eps,
                                               float* __restrict__ hout,
                                               int N, int layer, int do_relu) {
    __shared__ float zbuf[HD * LSTR];        // 67584 B : activations (z, then t)
    __shared__ float wbuf[(HD / 2) * WSTR];  // 73728 B : packed W1 then W2

    const int tid  = (int)threadIdx.x;
    const int w    = tid >> 5;
    const int lane = tid & 31;
    const int half = lane >> 4;
    const int lid  = lane & 15;
    const int rb   = (int)blockIdx.x * HD;

    const float s = 1.0f + eps[layer];
    const float* W1l = W1 + (size_t)layer * HD * HD;
    const float* W2l = W2 + (size_t)layer * HD * HD;
    const float* b1l = b1 + (size_t)layer * HD;
    const float* b2l = b2 + (size_t)layer * HD;

    // cooperative load: z tile (clamped tail rows)
    for (int i = tid; i < HD * (HD / 4); i += 256) {
        int r = i >> 5, c4 = i & 31;
        int gr = rb + r;
        int grc = gr < N ? gr : N - 1;
        float4 hv = ((const float4*)hin)[(size_t)grc * (HD / 4) + c4];
        float4 av = ((const float4*)agg)[(size_t)grc * (HD / 4) + c4];
        float4 z;
        z.x = s * hv.x + av.x;  z.y = s * hv.y + av.y;
        z.z = s * hv.z + av.z;  z.w = s * hv.w + av.w;
        *(float4*)&zbuf[(size_t)r * LSTR + c4 * 4] = z;
    }
    load_weights_packed(W1l, wbuf, tid);
    __syncthreads();

    // GEMM1: t-strip = ReLU(z @ W1 + b1)
    v8f acc[8];
#pragma unroll
    for (int n = 0; n < 8; ++n) acc[n] = (v8f){0,0,0,0,0,0,0,0};
    gemm_strip(zbuf, wbuf, w * 16 + lid, half, lid, acc);

    // epilogue 1: bias + ReLU, write t back into own zbuf strip
#pragma unroll
    for (int n = 0; n < 8; ++n) {
        float bias = b1l[n * 16 + lid];
#pragma unroll
        for (int v = 0; v < 8; ++v) {
            float val = acc[n][v] + bias;
            zbuf[(size_t)(w * 16 + v + 8 * half) * LSTR + n * 16 + lid] =
                fmaxf(val, 0.0f);
        }
    }
    __syncthreads();

    // swap weights: W2 into packed wbuf
    load_weights_packed(W2l, wbuf, tid);
    __syncthreads();

    // GEMM2: out-strip = t @ W2 + b2 (+ ReLU between layers)
#pragma unroll
    for (int n = 0; n < 8; ++n) acc[n] = (v8f){0,0,0,0,0,0,0,0};
    gemm_strip(zbuf, wbuf, w * 16 + lid, half, lid, acc);

#pragma unroll
    for (int n = 0; n < 8; ++n) {
        float bias = b2l[n * 16 + lid];
#pragma unroll
        for (int v = 0; v < 8; ++v) {
            int grow = rb + w * 16 + v + 8 * half;
            if (grow < N) {
                float val = acc[n][v] + bias;
                if (do_relu) val = fmaxf(val, 0.0f);
                hout[(size_t)grow * HD + n * 16 + lid] = val;
            }
        }
    }
}

// ---------------------------------------------------------------------------
// Per-graph mean-pool accumulation: pooled[batch[n]] += h[n]; cnt[batch[n]] += 1
// ---------------------------------------------------------------------------
__global__ __launch_bounds__(256) void gin_pool(const float* __restrict__ h,
                                                const int* __restrict__ batch,
                                                float* __restrict__ pooled,
                                                float* __restrict__ cnt,
                                                int N) {
    int gw   = (int)((blockIdx.x * blockDim.x + threadIdx.x) >> 5);
    int lane = (int)(threadIdx.x & 31);
    int nw   = (int)((gridDim.x * blockDim.x) >> 5);
    for (int n = gw; n < N; n += nw) {
        int g = batch[n];
        float4 v = ((const float4*)h)[(size_t)n * (HD / 4) + lane];
        float* p = pooled + (size_t)g * HD + lane * 4;
        unsafeAtomicAdd(p + 0, v.x);
        unsafeAtomicAdd(p + 1, v.y);
        unsafeAtomicAdd(p + 2, v.z);
        unsafeAtomicAdd(p + 3, v.w);
        if (lane == 0) unsafeAtomicAdd(&cnt[g], 1.0f);
    }
}

// ---------------------------------------------------------------------------
// out[g] = dot(pooled[g], Wh) / max(cnt[g],1) + bh     (one wave per graph)
// ---------------------------------------------------------------------------
__global__ __launch_bounds__(256) void gin_final(const float* __restrict__ pooled,
                                                 const float* __restrict__ cnt,
                                                 const float* __restrict__ Wh,
                                                 const float* __restrict__ bh,
                                                 float* __restrict__ out,
                                                 int G) {
    int gw   = (int)((blockIdx.x * blockDim.x + threadIdx.x) >> 5);
    int lane = (int)(threadIdx.x & 31);
    if (gw >= G) return;
    const float* p = pooled + (size_t)gw * HD;
    float a = 0.0f;
#pragma unroll
    for (int i = 0; i < 4; ++i) {
        int c = lane * 4 + i;
        a += p[c] * Wh[c];
    }
    for (int off = 16; off; off >>= 1) a += __shfl_down(a, off);
    if (lane == 0) {
        float c = cnt[gw];
        c = c > 1.0f ? c : 1.0f;
        out[gw] = a / c + bh[0];
    }
}

// ---------------------------------------------------------------------------
extern "C" void kernel_launch(void* const* d_in, const int* in_sizes, int n_in,
                              void* d_out, int out_size, void* d_ws, size_t ws_size,
                              hipStream_t stream) {
    const float* x     = (const float*)d_in[0];
    const int*   edge  = (const int*)d_in[1];
    const int*   batch = (const int*)d_in[2];
    const float* W1    = (const float*)d_in[3];
    const float* b1    = (const float*)d_in[4];
    const float* W2    = (const float*)d_in[5];
    const float* b2    = (const float*)d_in[6];
    const float* eps   = (const float*)d_in[7];
    const float* Wh    = (const float*)d_in[8];
    const float* bh    = (const float*)d_in[9];
    float* out = (float*)d_out;

    const int N = in_sizes[2];          // batch has N elements
    const int E = in_sizes[1] / 2;      // edge_index is [2, E]
    const int L = in_sizes[7];          // eps has L elements

    const int* src = edge;
    const int* dst = edge + E;

    size_t Nc = (size_t)N * HD;
    float* f0     = (float*)d_ws;
    float* f1     = f0 + Nc;
    float* agg    = f1 + Nc;
    float* pooled = agg + Nc;
    float* cnt    = pooled + (size_t)NGRAPH * HD;

    const int mlpBlocks  = (N + HD - 1) / HD;
    const int scatBlocks = 1024;   // 8192 waves grid-stride over 800K edges
    const int poolBlocks = 512;

    const float* hprev = x;
    float* bufs[2] = {f0, f1};
    for (int l = 0; l < L; ++l) {
        float* hnext = bufs[l & 1];
        hipMemsetAsync(agg, 0, Nc * sizeof(float), stream);
        gin_scatter<<<scatBlocks, 256, 0, stream>>>(hprev, src, dst, agg, E);
        gin_mlp<<<mlpBlocks, 256, 0, stream>>>(hprev, agg, W1, b1, W2, b2, eps,
                                               hnext, N, l, (l < L - 1) ? 1 : 0);
        hprev = hnext;
    }

    hipMemsetAsync(pooled, 0, (size_t)NGRAPH * HD * sizeof(float), stream);
    hipMemsetAsync(cnt, 0, NGRAPH * sizeof(float), stream);
    gin_pool<<<poolBlocks, 256, 0, stream>>>(hprev, batch, pooled, cnt, N);
    gin_final<<<(NGRAPH * 32 + 255) / 256, 256, 0, stream>>>(pooled, cnt, Wh, bh,
                                                             out, NGRAPH);
}